// RefLocal8_53257594470977
// MI455X (gfx1250) — compile-verified
//
#include <hip/hip_runtime.h>

typedef __attribute__((ext_vector_type(2))) float v2f;
typedef __attribute__((ext_vector_type(8))) float v8f;

#define BB 4
#define HH 128
#define WW 128
#define CC 64
#define DD 64
#define NPIX (BB * HH * WW)

#define TY 8
#define TX 16
#define LROWS (TY + 4) /* 12 */
#define LCOLS (TX + 4) /* 20 */
#define PITCH 68       /* floats per pixel row in LDS: 272B, 16B aligned, bank-rotating */

// ---------------------------------------------------------------------------
// Kernel 1: M2[c', c] = sum_d w_ref[c',d] * w_main[c,d]   (64x64 = W_ref @ W_main^T)
// One workgroup, 4 waves; each wave owns a 16-row stripe, loops over 4 col tiles.
// ---------------------------------------------------------------------------
__global__ void fuse_weights_kernel(const float* __restrict__ w_ref,
                                    const float* __restrict__ w_main,
                                    float* __restrict__ M2) {
  const int wave = threadIdx.x >> 5;
  const int lane = threadIdx.x & 31;
  const int half = lane >> 4;   // 0 | 1
  const int l16  = lane & 15;
  const int kb   = half << 1;   // A/B K-lane offset: 0 | 2
  const int mrow = (wave << 4) + l16;

  for (int ct = 0; ct < 4; ++ct) {
    const int c = (ct << 4) + l16;
    v8f acc = {0.f, 0.f, 0.f, 0.f, 0.f, 0.f, 0.f, 0.f};
    #pragma unroll
    for (int k = 0; k < CC; k += 4) {
      // A = w_ref tile: lane holds rows m, K = {k+kb, k+kb+1}
      v2f a = *(const v2f*)(w_ref + mrow * DD + k + kb);
      // B[d][c] = w_main[c][d]: consecutive d -> contiguous float2
      v2f b = *(const v2f*)(w_main + c * DD + k + kb);
      acc = __builtin_amdgcn_wmma_f32_16x16x4_f32(false, a, false, b,
                                                  (short)0, acc, false, false);
    }
    const int row0 = (wave << 4) + (half << 3);
    #pragma unroll
    for (int v = 0; v < 8; ++v)
      M2[(row0 + v) * CC + (ct << 4) + l16] = acc[v];
  }
}

// ---------------------------------------------------------------------------
// Kernel 2: G = ref @ M2    ([NPIX,64] x [64,64] -> [NPIX,64])
// One wave per 16(pixel) x 16(channel) output tile; K=64 via 16 chained
// v_wmma_f32_16x16x4_f32.
// ---------------------------------------------------------------------------
__global__ void conv_g_kernel(const float* __restrict__ ref,
                              const float* __restrict__ M2,
                              float* __restrict__ G) {
  const int gwave = (int)((blockIdx.x * blockDim.x + threadIdx.x) >> 5);
  const int lane  = threadIdx.x & 31;
  const int half  = lane >> 4;
  const int l16   = lane & 15;
  const int kb    = half << 1;
  const int mt    = gwave >> 2;  // pixel tile
  const int nt    = gwave & 3;   // channel tile
  const int m     = (mt << 4) + l16;
  const int n     = (nt << 4) + l16;

  const float* arow = ref + (size_t)m * CC;
  v8f acc = {0.f, 0.f, 0.f, 0.f, 0.f, 0.f, 0.f, 0.f};
  #pragma unroll
  for (int k = 0; k < CC; k += 4) {
    v2f a = *(const v2f*)(arow + k + kb);   // ref[m][k+kb .. k+kb+1]
    v2f b;
    b.x = M2[(k + kb) * DD + n];            // M2[d][n], rows d, d+1
    b.y = M2[(k + kb + 1) * DD + n];
    acc = __builtin_amdgcn_wmma_f32_16x16x4_f32(false, a, false, b,
                                                (short)0, acc, false, false);
  }
  const int row0 = (mt << 4) + (half << 3);
  #pragma unroll
  for (int v = 0; v < 8; ++v)
    G[(size_t)(row0 + v) * DD + (nt << 4) + l16] = acc[v];
}

// ---------------------------------------------------------------------------
// Kernel 3: attn[p,k] = dot(main[p], G[p+off_k]) over D=64, softmax over the
// 5x5 window (zero-padded neighbors contribute score exactly 0, as in ref).
// 16x8 pixel tile per block; 20x12 halo tile of G staged in LDS (pitch 68).
// ---------------------------------------------------------------------------
__global__ void attn_softmax_kernel(const float* __restrict__ mainp,
                                    const float* __restrict__ G,
                                    float* __restrict__ out) {
  extern __shared__ float sG[];
  const int tx  = threadIdx.x, ty = threadIdx.y;
  const int tid = ty * TX + tx;
  const int x0  = blockIdx.x * TX;
  const int y0  = blockIdx.y * TY;
  const int b   = blockIdx.z;

  // Cooperative zero-padded load of the G halo tile into LDS (float4 chunks).
  const int total = LROWS * LCOLS * (DD / 4);
  for (int i = tid; i < total; i += TX * TY) {
    const int c4 = i & 15;
    const int pi = i >> 4;
    const int yy = pi / LCOLS;
    const int xx = pi - yy * LCOLS;
    const int gy = y0 + yy - 2;
    const int gx = x0 + xx - 2;
    float4 v = make_float4(0.f, 0.f, 0.f, 0.f);
    if (gy >= 0 && gy < HH && gx >= 0 && gx < WW)
      v = *(const float4*)(G + ((((size_t)b * HH + gy) * WW + gx) << 6) + (c4 << 2));
    *(float4*)(&sG[pi * PITCH + (c4 << 2)]) = v;
  }
  __syncthreads();

  // Query vector = raw main[p] (conv folded into G).
  const int y = y0 + ty, x = x0 + tx;
  const float* mrow = mainp + ((((size_t)b * HH + y) * WW + x) << 6);
  float4 q[16];
  #pragma unroll
  for (int i = 0; i < 16; ++i) q[i] = *(const float4*)(mrow + (i << 2));

  float sc[25];
  #pragma unroll
  for (int ki = 0; ki < 5; ++ki) {
    #pragma unroll
    for (int kj = 0; kj < 5; ++kj) {
      const float* g = &sG[((ty + ki) * LCOLS + (tx + kj)) * PITCH];
      float acc = 0.f;
      #pragma unroll
      for (int i = 0; i < 16; ++i) {
        float4 gv = *(const float4*)(g + (i << 2));
        acc = fmaf(q[i].x, gv.x, acc);
        acc = fmaf(q[i].y, gv.y, acc);
        acc = fmaf(q[i].z, gv.z, acc);
        acc = fmaf(q[i].w, gv.w, acc);
      }
      sc[ki * 5 + kj] = acc;
    }
  }

  // Softmax over the 25 window scores.
  float mx = sc[0];
  #pragma unroll
  for (int k = 1; k < 25; ++k) mx = fmaxf(mx, sc[k]);
  float sum = 0.f;
  #pragma unroll
  for (int k = 0; k < 25; ++k) {
    const float e = __expf(sc[k] - mx);
    sc[k] = e;
    sum += e;
  }
  const float inv = 1.f / sum;
  float* op = out + (((size_t)b * HH + y) * WW + x) * 25;
  #pragma unroll
  for (int k = 0; k < 25; ++k) op[k] = sc[k] * inv;
}

// ---------------------------------------------------------------------------
extern "C" void kernel_launch(void* const* d_in, const int* in_sizes, int n_in,
                              void* d_out, int out_size, void* d_ws, size_t ws_size,
                              hipStream_t stream) {
  const float* main_in = (const float*)d_in[0];
  const float* ref_in  = (const float*)d_in[1];
  const float* w_main  = (const float*)d_in[2];
  const float* w_ref   = (const float*)d_in[3];
  float* out = (float*)d_out;

  // Workspace layout: M2 (16 KB) | G (NPIX*64 f32 = 16.78 MB)
  float* M2 = (float*)d_ws;
  float* G  = (float*)((char*)d_ws + (size_t)CC * DD * sizeof(float));

  fuse_weights_kernel<<<1, 128, 0, stream>>>(w_ref, w_main, M2);

  // (NPIX/16) pixel tiles x 4 channel tiles = 16384 waves; 8 waves/block.
  const int nWaves  = (NPIX / 16) * 4;
  const int nBlocks = nWaves * 32 / 256;
  conv_g_kernel<<<nBlocks, 256, 0, stream>>>(ref_in, M2, G);

  dim3 grid(WW / TX, HH / TY, BB);
  dim3 blk(TX, TY);
  const size_t smem = (size_t)LROWS * LCOLS * PITCH * sizeof(float); // 65,280 B
  attn_softmax_kernel<<<grid, blk, smem, stream>>>(main_in, G, out);

  (void)in_sizes; (void)n_in; (void)out_size; (void)ws_size;
}